// Aggregator_60078002536942
// MI455X (gfx1250) — compile-verified
//
#include <hip/hip_runtime.h>
#include <math.h>

// ---------------------------------------------------------------------------
// KGAT-style aggregator for MI455X (gfx1250, wave32).
// Memory-bound scatter/gather done with float4 rows + global f32 atomics;
// the dense user-score GEMM uses V_WMMA_F32_16X16X4_F32 (f32 in/out, exact
// precision match to the f32 reference; score GEMM is <1% of total traffic).
// ---------------------------------------------------------------------------

static constexpr int kD  = 128;        // embedding dim
static constexpr int kD4 = kD / 4;     // float4 per row

typedef float v2f __attribute__((ext_vector_type(2)));
typedef float v8f __attribute__((ext_vector_type(8)));

// ---------------- init: zero outputs + counts ------------------------------
__global__ void init_zero_kernel(float* __restrict__ out, long out_n,
                                 float* __restrict__ cnt, int cnt_n) {
  long i = (long)blockIdx.x * blockDim.x + threadIdx.x;
  if (i < out_n) out[i] = 0.0f;
  if (i < cnt_n) cnt[i] = 0.0f;
}

// ---------------- KG edge scatter: one wave32 per edge ---------------------
__global__ void kg_edge_kernel(const float4* __restrict__ ent4,
                               const float4* __restrict__ rel4,
                               const int* __restrict__ edge_index,  // [2, E]
                               const int* __restrict__ edge_type,   // [E]
                               float* __restrict__ ent_out,         // [n_ent, 128]
                               float* __restrict__ cnt,             // [n_ent]
                               int n_edges) {
  int e    = (int)(((long)blockIdx.x * blockDim.x + threadIdx.x) >> 5);
  int lane = threadIdx.x & 31;
  if (e >= n_edges) return;

  int h = edge_index[e];
  int t = edge_index[n_edges + e];
  int r = edge_type[e];

  float4 ev = ent4[(long)t * kD4 + lane];   // coalesced 512B row gather
  float4 rv = rel4[(long)r * kD4 + lane];   // 32 rows total -> L2 resident

  float* dst = ent_out + (long)h * kD + lane * 4;
  atomicAdd(dst + 0, ev.x * rv.x);
  atomicAdd(dst + 1, ev.y * rv.y);
  atomicAdd(dst + 2, ev.z * rv.z);
  atomicAdd(dst + 3, ev.w * rv.w);
  if (lane == 0) atomicAdd(&cnt[h], 1.0f);
}

// ---------------- interact scatter: one wave32 per nnz ---------------------
__global__ void interact_kernel(const float4* __restrict__ ent4,
                                const int* __restrict__ rows,
                                const int* __restrict__ cols,
                                const float* __restrict__ vals,
                                float* __restrict__ user_out,  // [n_users, 128]
                                int nnz) {
  int e    = (int)(((long)blockIdx.x * blockDim.x + threadIdx.x) >> 5);
  int lane = threadIdx.x & 31;
  if (e >= nnz) return;

  int   u = rows[e];
  int   c = cols[e];
  float v = vals[e];

  float4 ev = ent4[(long)c * kD4 + lane];
  float* dst = user_out + (long)u * kD + lane * 4;
  atomicAdd(dst + 0, ev.x * v);
  atomicAdd(dst + 1, ev.y * v);
  atomicAdd(dst + 2, ev.z * v);
  atomicAdd(dst + 3, ev.w * v);
}

// ---------------- entity finalize: divide by max(count, 1) -----------------
__global__ void entity_final_kernel(float* __restrict__ ent_out,
                                    const float* __restrict__ cnt,
                                    long n_elem) {
  long i = (long)blockIdx.x * blockDim.x + threadIdx.x;
  if (i >= n_elem) return;
  float c = cnt[i >> 7];  // i / 128
  ent_out[i] = ent_out[i] / fmaxf(c, 1.0f);
}

// ---------------- disen_weight = softmax(att) @ relation_emb ---------------
// one block of 128 threads; n_fac=4, n_rel=32
__global__ void disen_kernel(const float* __restrict__ att,   // [4, 32]
                             const float* __restrict__ rel,   // [32, 128]
                             float* __restrict__ dw,          // [4, 128]
                             int n_fac, int n_rel) {
  __shared__ float w[4 * 32];
  int tid = threadIdx.x;  // 0..127
  if (tid < n_fac * n_rel) w[tid] = att[tid];
  __syncthreads();
  if (tid < n_fac) {
    float m = -1e30f;
    for (int r = 0; r < n_rel; ++r) m = fmaxf(m, w[tid * n_rel + r]);
    float s = 0.0f;
    for (int r = 0; r < n_rel; ++r) {
      float e = __expf(w[tid * n_rel + r] - m);
      w[tid * n_rel + r] = e;
      s += e;
    }
    float inv = 1.0f / s;
    for (int r = 0; r < n_rel; ++r) w[tid * n_rel + r] *= inv;
  }
  __syncthreads();
  int d = tid;  // one column per thread
  for (int f = 0; f < n_fac; ++f) {
    float acc = 0.0f;
    for (int r = 0; r < n_rel; ++r) acc += w[f * n_rel + r] * rel[r * kD + d];
    dw[f * kD + d] = acc;
  }
}

// ---------------- user finalize: WMMA score GEMM + softmax + gate ----------
// 128 threads = 4 wave32s per block; each wave owns a 16-user tile.
// score = softmax(U[16,128] @ L^T[128,4]) via V_WMMA_F32_16X16X4_F32
// (32 K-steps); then user_out = user_out * (1 + score @ disen_weight).
__global__ void __launch_bounds__(128)
user_score_kernel(const float4* __restrict__ user4,   // [n_users, 32] float4
                  const float4* __restrict__ latent4, // [4, 32] float4
                  const float4* __restrict__ dw4,     // [4, 32] float4
                  float4* __restrict__ user_out4,     // [n_users, 32] float4
                  int n_users) {
  __shared__ float4 As4[4 * 16 * kD4];  // 4 waves x (16 users x 128) = 32KB
  __shared__ float4 Ls4[4 * kD4];       // latent factors, 2KB
  __shared__ float4 Ds4[4 * kD4];       // disen weights, 2KB
  __shared__ float  Sc[4 * 16 * 4];     // per-wave per-user factor scores

  const int  tid  = threadIdx.x;
  const int  wid  = tid >> 5;          // wave 0..3
  const int  lane = tid & 31;
  const long u0   = ((long)blockIdx.x * 4 + wid) * 16;

  const float* As = (const float*)&As4[wid * 16 * kD4];
  const float* Ls = (const float*)Ls4;

  // Stage this wave's 16-user tile, float4-coalesced (512 float4 per wave).
  for (int i = lane; i < 16 * kD4; i += 32) {
    long row = u0 + (i >> 5);  // 32 float4 per row
    As4[wid * 16 * kD4 + i] =
        (row < (long)n_users) ? user4[row * kD4 + (i & 31)]
                              : make_float4(0.f, 0.f, 0.f, 0.f);
  }
  // Stage latent + disen weights once per block (128 float4 each).
  for (int i = tid; i < 4 * kD4; i += 128) {
    Ls4[i] = latent4[i];
    Ds4[i] = dw4[i];
  }
  __syncthreads();

  const int half = lane >> 4;   // 0: K={0,1}, 1: K={2,3}
  const int l16  = lane & 15;   // A: row M; B: col N

  // B columns n>=4 are zero: clamp the LDS index, scale by 0/1 mask.
  // (avoids exec-predicated ds_loads; becomes plain ds_load_b64 + v_mul)
  const float bmask = (l16 < 4) ? 1.0f : 0.0f;
  const int   brow  = (l16 < 4) ? l16 : 0;

  const float2* As2 = (const float2*)As;   // pairwise -> ds_load_b64
  const float2* Ls2 = (const float2*)Ls;

  v8f c = {};
  for (int k = 0; k < kD; k += 4) {
    // A 16x4 f32 frag: lane l16 holds row M=l16; (VGPR0,VGPR1) = K{0,1}/K{2,3}
    float2 ap = As2[l16 * (kD / 2) + (k >> 1) + half];
    // B 4x16 f32 frag: B[kk][n] = latent[n][k+kk] (n<4), zero otherwise
    float2 bp = Ls2[brow * (kD / 2) + (k >> 1) + half];
    v2f a, b;
    a.x = ap.x;         a.y = ap.y;
    b.x = bp.x * bmask; b.y = bp.y * bmask;
    c = __builtin_amdgcn_wmma_f32_16x16x4_f32(
        /*neg_a=*/false, a, /*neg_b=*/false, b,
        /*c_mod=*/(short)0, c, /*reuse_a=*/false, /*reuse_b=*/false);
  }

  // C layout: VGPR v of lane holds row M = v + 8*half, col N = l16.
  if (l16 < 4) {
    for (int v = 0; v < 8; ++v)
      Sc[(wid * 16 + v + 8 * half) * 4 + l16] = c[v];
  }
  __syncthreads();

  // Per-user softmax over the 4 factor logits (lanes 0..15, one user each).
  if (lane < 16) {
    int s = (wid * 16 + lane) * 4;
    float s0 = Sc[s + 0], s1 = Sc[s + 1], s2 = Sc[s + 2], s3 = Sc[s + 3];
    float m  = fmaxf(fmaxf(s0, s1), fmaxf(s2, s3));
    float e0 = __expf(s0 - m), e1 = __expf(s1 - m);
    float e2 = __expf(s2 - m), e3 = __expf(s3 - m);
    float inv = 1.0f / (e0 + e1 + e2 + e3);
    Sc[s + 0] = e0 * inv;
    Sc[s + 1] = e1 * inv;
    Sc[s + 2] = e2 * inv;
    Sc[s + 3] = e3 * inv;
  }
  __syncthreads();

  // Gated in-place float4 update: one float4 per lane per user row.
  for (int m = 0; m < 16; ++m) {
    long row = u0 + m;
    if (row >= (long)n_users) break;
    int s = (wid * 16 + m) * 4;
    float f0 = Sc[s + 0], f1 = Sc[s + 1], f2 = Sc[s + 2], f3 = Sc[s + 3];
    float4 d0 = Ds4[0 * kD4 + lane];
    float4 d1 = Ds4[1 * kD4 + lane];
    float4 d2 = Ds4[2 * kD4 + lane];
    float4 d3 = Ds4[3 * kD4 + lane];
    float4 g;
    g.x = 1.0f + f0 * d0.x + f1 * d1.x + f2 * d2.x + f3 * d3.x;
    g.y = 1.0f + f0 * d0.y + f1 * d1.y + f2 * d2.y + f3 * d3.y;
    g.z = 1.0f + f0 * d0.z + f1 * d1.z + f2 * d2.z + f3 * d3.z;
    g.w = 1.0f + f0 * d0.w + f1 * d1.w + f2 * d2.w + f3 * d3.w;
    long idx = row * kD4 + lane;
    float4 u = user_out4[idx];
    u.x *= g.x; u.y *= g.y; u.z *= g.z; u.w *= g.w;
    user_out4[idx] = u;
  }
}

// ---------------------------------------------------------------------------
extern "C" void kernel_launch(void* const* d_in, const int* in_sizes, int n_in,
                              void* d_out, int out_size, void* d_ws, size_t ws_size,
                              hipStream_t stream) {
  const float* entity_emb   = (const float*)d_in[0];
  const float* user_emb     = (const float*)d_in[1];
  const float* latent_emb   = (const float*)d_in[2];
  const float* relation_emb = (const float*)d_in[3];
  const int*   edge_index   = (const int*)d_in[4];
  const int*   edge_type    = (const int*)d_in[5];
  const int*   inter_rows   = (const int*)d_in[6];
  const int*   inter_cols   = (const int*)d_in[7];
  const float* inter_vals   = (const float*)d_in[8];
  const float* disen_att    = (const float*)d_in[9];

  const int n_entities = in_sizes[0] / kD;
  const int n_users    = in_sizes[1] / kD;
  const int n_factors  = in_sizes[2] / kD;   // 4
  const int n_rel      = in_sizes[3] / kD;   // 32
  const int n_edges    = in_sizes[5];
  const int nnz        = in_sizes[8];

  float* ent_out  = (float*)d_out;
  float* user_out = ent_out + (size_t)n_entities * kD;

  // workspace layout: [counts: n_entities floats][dw: 4*128 floats]
  // (n_entities*4 bytes is 16B-aligned for n_entities % 4 == 0)
  float* cnt = (float*)d_ws;
  float* dw  = cnt + n_entities;

  const int T = 256;

  // 1) zero outputs + counts
  {
    long total  = (long)out_size;
    long blocks = (total + T - 1) / T;
    init_zero_kernel<<<(int)blocks, T, 0, stream>>>(ent_out, total, cnt, n_entities);
  }

  // 2) KG edge scatter-add (8 edges per 256-thread block, wave32 per edge)
  {
    long blocks = ((long)n_edges + 7) / 8;
    kg_edge_kernel<<<(int)blocks, T, 0, stream>>>(
        (const float4*)entity_emb, (const float4*)relation_emb,
        edge_index, edge_type, ent_out, cnt, n_edges);
  }

  // 3) interact scatter-add
  {
    long blocks = ((long)nnz + 7) / 8;
    interact_kernel<<<(int)blocks, T, 0, stream>>>(
        (const float4*)entity_emb, inter_rows, inter_cols, inter_vals,
        user_out, nnz);
  }

  // 4) disen_weight = softmax(att) @ relation_emb
  disen_kernel<<<1, 128, 0, stream>>>(disen_att, relation_emb, dw, n_factors, n_rel);

  // 5) entity mean
  {
    long n_elem = (long)n_entities * kD;
    long blocks = (n_elem + T - 1) / T;
    entity_final_kernel<<<(int)blocks, T, 0, stream>>>(ent_out, cnt, n_elem);
  }

  // 6) WMMA user score + softmax + gated in-place finalize (64 users/block)
  {
    int blocks = (n_users + 63) / 64;
    user_score_kernel<<<blocks, 128, 0, stream>>>(
        (const float4*)user_emb, (const float4*)latent_emb,
        (const float4*)dw, (float4*)user_out, n_users);
  }
}